// CountEmbedder_45286135169615
// MI455X (gfx1250) — compile-verified
//
#include <hip/hip_runtime.h>
#include <hip/hip_bf16.h>
#include <stdint.h>

// ---------------------------------------------------------------------------
// Bag-of-words counts on MI455X (gfx1250).
//  - One workgroup per document row.
//  - Full vocab histogram lives in LDS (50257 x u32 = ~196 KB; CDNA5 WGP has
//    320 KB LDS, so this fits -- impossible on CDNA4's 64 KB).
//  - Tokens + mask for the row are copied global->LDS with the gfx1250 async
//    LDS-DMA path (ASYNCcnt), overlapped with zeroing the histogram.
//  - Counting uses DS atomics (ds_add_u32) -- zero global atomics.
//  - Output (206 MB, write-once) streamed with non-temporal b128 stores.
// ---------------------------------------------------------------------------

#define VOCAB    50257
#define VPAD     50260            // pad to multiple of 4 for clean zeroing
#define SEQ      2048
#define NTHREADS 1024

typedef float v4f __attribute__((ext_vector_type(4)));
typedef int   v4i __attribute__((ext_vector_type(4)));
typedef __attribute__((address_space(1))) v4i* g4_ptr;   // global int4*
typedef __attribute__((address_space(3))) v4i* l4_ptr;   // LDS int4*

#if defined(__gfx1250__) && \
    __has_builtin(__builtin_amdgcn_global_load_async_to_lds_b128)
#define USE_ASYNC_LDS 1
#else
#define USE_ASYNC_LDS 0
#endif

__device__ __forceinline__ void wait_async_zero() {
#if __has_builtin(__builtin_amdgcn_s_wait_asynccnt)
  __builtin_amdgcn_s_wait_asynccnt(0);
#else
  asm volatile("s_wait_asynccnt 0" ::: "memory");
#endif
}

__global__ __launch_bounds__(NTHREADS) void CountEmbedder_bow_kernel(
    const int* __restrict__ token_ids,     // [B, SEQ] int32
    const uint8_t* __restrict__ mask,      // [B, SEQ] bool (1 byte)
    float* __restrict__ out)               // [B, VOCAB] f32
{
  __shared__ __align__(16) unsigned int hist[VPAD];   // 201,040 B
  __shared__ __align__(16) int          tok[SEQ];     //   8,192 B
  __shared__ __align__(16) uint8_t      msk[SEQ];     //   2,048 B

  const int b   = blockIdx.x;
  const int tid = threadIdx.x;

  const int*     tok_row = token_ids + (size_t)b * SEQ;
  const uint8_t* msk_row = mask      + (size_t)b * SEQ;

  // ---- Phase A: launch async global->LDS copies of this row's inputs ----
#if USE_ASYNC_LDS
  if (tid < SEQ / 4) {            // 512 lanes x 16 B = 8 KB of tokens
    __builtin_amdgcn_global_load_async_to_lds_b128(
        (g4_ptr)(void*)(tok_row + tid * 4),
        (l4_ptr)(void*)(&tok[tid * 4]),
        /*offset=*/0, /*cpol=*/0);
  }
  if (tid < SEQ / 16) {           // 128 lanes x 16 B = 2 KB of mask bytes
    __builtin_amdgcn_global_load_async_to_lds_b128(
        (g4_ptr)(void*)(msk_row + tid * 16),
        (l4_ptr)(void*)(&msk[tid * 16]),
        /*offset=*/0, /*cpol=*/0);
  }
#else
  for (int s = tid; s < SEQ; s += NTHREADS) {
    tok[s] = tok_row[s];
    msk[s] = msk_row[s];
  }
#endif

  // ---- Phase B: zero the LDS histogram while the DMA is in flight ----
  for (int v = tid; v < VPAD; v += NTHREADS) hist[v] = 0u;

#if USE_ASYNC_LDS
  wait_async_zero();
#endif
  __syncthreads();

  // ---- Phase C: per-row histogram via LDS atomics (ds_add_u32) ----
  for (int s = tid; s < SEQ; s += NTHREADS) {        // 2 iterations
    if (msk[s]) atomicAdd(&hist[tok[s]], 1u);
  }
  __syncthreads();

  // ---- Phase D: stream u32 counts to global as f32, non-temporal ----
  float* row = out + (size_t)b * VOCAB;

  // Row stride is 50257*4 B => only 4 B aligned; peel to 16 B alignment.
  int head = (int)(((16u - ((uint32_t)(uintptr_t)row & 15u)) & 15u) >> 2);
  if (head > VOCAB) head = VOCAB;
  if (tid < head) {
    __builtin_nontemporal_store((float)hist[tid], row + tid);
  }

  const int nv4 = (VOCAB - head) >> 2;               // 16 B-aligned body
  for (int i = tid; i < nv4; i += NTHREADS) {
    const int j = head + i * 4;
    v4f val = { (float)hist[j + 0], (float)hist[j + 1],
                (float)hist[j + 2], (float)hist[j + 3] };
    __builtin_nontemporal_store(val, (v4f*)(row + j));
  }

  for (int v = head + nv4 * 4 + tid; v < VOCAB; v += NTHREADS) {
    __builtin_nontemporal_store((float)hist[v], row + v);   // tail
  }
}

extern "C" void kernel_launch(void* const* d_in, const int* in_sizes, int n_in,
                              void* d_out, int out_size, void* d_ws, size_t ws_size,
                              hipStream_t stream) {
  const int*     token_ids = (const int*)d_in[0];
  const uint8_t* mask      = (const uint8_t*)d_in[1];
  float*         out       = (float*)d_out;

  const int B = in_sizes[0] / SEQ;   // 1024 for the reference shapes

  CountEmbedder_bow_kernel<<<dim3(B), dim3(NTHREADS), 0, stream>>>(
      token_ids, mask, out);
}